// MolecularProdRuleEmbedding_5076651344547
// MI455X (gfx1250) — compile-verified
//
#include <hip/hip_runtime.h>
#include <hip/hip_bf16.h>

typedef __attribute__((ext_vector_type(2))) float v2f;
typedef __attribute__((ext_vector_type(8))) float v8f;

#define NRULES 4096
#define DIM 32

__device__ __forceinline__ void atomAddF32(float* p, float v) {
    unsafeAtomicAdd(p, v);   // lowers to global_atomic_add_f32 (no CAS loop)
}

// ---------------------------------------------------------------------------
// Kernel 1: initial embeddings  h_e = atom[edge_sym],
//           h_n = bond[node_sym] + has_ext * ext[node_ext];  also zero fp.
// ---------------------------------------------------------------------------
__global__ __launch_bounds__(256) void init_kernel(
    float* __restrict__ h_e, float* __restrict__ h_n, float* __restrict__ fp,
    const float* __restrict__ atom_embed, const float* __restrict__ bond_embed,
    const float* __restrict__ ext_embed,
    const int* __restrict__ edge_sym, const int* __restrict__ node_sym,
    const int* __restrict__ node_ext, const float* __restrict__ node_has_ext,
    int E, int N)
{
    long t = (long)blockIdx.x * blockDim.x + threadIdx.x;
    long eEnd = (long)E * DIM;
    long nEnd = eEnd + (long)N * DIM;
    long fEnd = nEnd + (long)NRULES * DIM;
    if (t < eEnd) {
        int e = (int)(t >> 5), d = (int)(t & 31);
        h_e[t] = atom_embed[edge_sym[e] * DIM + d];
    } else if (t < nEnd) {
        long u = t - eEnd;
        int n = (int)(u >> 5), d = (int)(u & 31);
        h_n[u] = bond_embed[node_sym[n] * DIM + d] +
                 node_has_ext[n] * ext_embed[node_ext[n] * DIM + d];
    } else if (t < fEnd) {
        fp[t - nEnd] = 0.0f;
    }
}

// ---------------------------------------------------------------------------
// Kernel 2: seed accumulators  v_e = h_e ; v_n = h_n  (float4 wide)
// ---------------------------------------------------------------------------
__global__ __launch_bounds__(256) void seed_kernel(
    float4* __restrict__ v_e, const float4* __restrict__ h_e,
    float4* __restrict__ v_n, const float4* __restrict__ h_n,
    int E, int N)
{
    long t = (long)blockIdx.x * blockDim.x + threadIdx.x;   // (E+N)*8 float4s
    long eEnd = (long)E * (DIM / 4);
    long nEnd = eEnd + (long)N * (DIM / 4);
    if (t < eEnd)       v_e[t] = h_e[t];
    else if (t < nEnd)  v_n[t - eEnd] = h_n[t - eEnd];
}

// ---------------------------------------------------------------------------
// Kernel 3: incidence scatter:
//   v_e[inc_edge[i]] += h_n[inc_node[i]] ; v_n[inc_node[i]] += h_e[inc_edge[i]]
// ---------------------------------------------------------------------------
__global__ __launch_bounds__(256) void scatter_kernel(
    float* __restrict__ v_e, float* __restrict__ v_n,
    const float* __restrict__ h_e, const float* __restrict__ h_n,
    const int* __restrict__ inc_edge, const int* __restrict__ inc_node,
    int I2)
{
    long t = (long)blockIdx.x * blockDim.x + threadIdx.x;   // I2*32 threads
    if (t >= (long)I2 * DIM) return;
    int i = (int)(t >> 5), d = (int)(t & 31);
    int e = inc_edge[i], n = inc_node[i];
    atomAddF32(&v_e[(long)e * DIM + d], h_n[(long)n * DIM + d]);
    atomAddF32(&v_n[(long)n * DIM + d], h_e[(long)e * DIM + d]);
}

// ---------------------------------------------------------------------------
// Kernel 4: per-tile dual GEMM via V_WMMA_F32_16X16X4_F32.
//   One wave owns 16 rows of v[R,32]; computes
//     h_out = relu(v @ Wl + bl)           (written back)
//     fp[rule[row]] += relu(v @ Wo + bo)  (atomic accumulate)
//   K=32 -> 8 chunks of K=4; N=32 -> two 16-wide accumulators per matrix.
//   A frag (16x4 f32, ISA 7.12.2): lanes 0-15 hold K={0,1}, lanes 16-31 K={2,3}.
//   B frag (4x16 f32): N across lanes, same K split across lane halves.
// ---------------------------------------------------------------------------
__global__ __launch_bounds__(128) void tile_gemm_kernel(
    const float* __restrict__ v, float* __restrict__ h_out,
    float* __restrict__ fp, const int* __restrict__ rule,
    const float* __restrict__ Wl, const float* __restrict__ bl,
    const float* __restrict__ Wo, const float* __restrict__ bo,
    int R)
{
    const int lane = threadIdx.x & 31;
    const int wave = threadIdx.x >> 5;
    const int tile = blockIdx.x * 4 + wave;
    const int base = tile * 16;
    if (base >= R) return;              // wave-uniform: EXEC stays all-1s

    const int hi = lane >> 4;           // lane half selects K sub-pair
    const int m  = lane & 15;           // row within tile (A) / column (B,C,D)
    const int n  = m;
    const int row = min(base + m, R - 1);
    const float* __restrict__ arow = v + (long)row * DIM;

    // Preload B fragments for both weight matrices, both N-halves.
    v2f Bl0[8], Bl1[8], Bo0[8], Bo1[8];
#pragma unroll
    for (int k = 0; k < 8; ++k) {
        int kr = 4 * k + 2 * hi;
        Bl0[k] = (v2f){ Wl[kr * DIM + n],      Wl[(kr + 1) * DIM + n]      };
        Bl1[k] = (v2f){ Wl[kr * DIM + 16 + n], Wl[(kr + 1) * DIM + 16 + n] };
        Bo0[k] = (v2f){ Wo[kr * DIM + n],      Wo[(kr + 1) * DIM + n]      };
        Bo1[k] = (v2f){ Wo[kr * DIM + 16 + n], Wo[(kr + 1) * DIM + 16 + n] };
    }

    v8f cl0 = {}, cl1 = {}, co0 = {}, co1 = {};
#pragma unroll
    for (int k = 0; k < 8; ++k) {
        int kk = 4 * k + 2 * hi;
        v2f a = (v2f){ arow[kk], arow[kk + 1] };
        cl0 = __builtin_amdgcn_wmma_f32_16x16x4_f32(false, a, false, Bl0[k],
                                                    (short)0, cl0, false, false);
        cl1 = __builtin_amdgcn_wmma_f32_16x16x4_f32(false, a, false, Bl1[k],
                                                    (short)0, cl1, false, false);
        co0 = __builtin_amdgcn_wmma_f32_16x16x4_f32(false, a, false, Bo0[k],
                                                    (short)0, co0, false, false);
        co1 = __builtin_amdgcn_wmma_f32_16x16x4_f32(false, a, false, Bo1[k],
                                                    (short)0, co1, false, false);
    }

    const float bl0v = bl[n], bl1v = bl[16 + n];
    const float bo0v = bo[n], bo1v = bo[16 + n];

    // C/D layout (16x16 f32): VGPR r, lanes 0-15 -> M=r, lanes 16-31 -> M=8+r.
#pragma unroll
    for (int r = 0; r < 8; ++r) {
        int orow = base + r + 8 * hi;
        if (orow < R) {
            float xl0 = fmaxf(cl0[r] + bl0v, 0.0f);
            float xl1 = fmaxf(cl1[r] + bl1v, 0.0f);
            h_out[(long)orow * DIM + n]      = xl0;
            h_out[(long)orow * DIM + 16 + n] = xl1;
            int rr = rule[orow];
            float xo0 = fmaxf(co0[r] + bo0v, 0.0f);
            float xo1 = fmaxf(co1[r] + bo1v, 0.0f);
            atomAddF32(&fp[(long)rr * DIM + n],      xo0);
            atomAddF32(&fp[(long)rr * DIM + 16 + n], xo1);
        }
    }
}

// ---------------------------------------------------------------------------
// Kernel 5: output gather  out[i] = idx<NRULES ? fp[idx] : 0   (float4 wide)
// ---------------------------------------------------------------------------
__global__ __launch_bounds__(256) void gather_kernel(
    const int* __restrict__ seq, const float4* __restrict__ fp4,
    float4* __restrict__ out4, long total)
{
    long t = (long)blockIdx.x * blockDim.x + threadIdx.x;   // total * 8
    if (t >= total * (DIM / 4)) return;
    long i = t >> 3;
    int  q = (int)(t & 7);
    int idx = seq[i];
    float4 val;
    if (idx < NRULES) {
        val = fp4[(long)idx * (DIM / 4) + q];
    } else {
        val = make_float4(0.0f, 0.0f, 0.0f, 0.0f);
    }
    out4[t] = val;
}

// ---------------------------------------------------------------------------
extern "C" void kernel_launch(void* const* d_in, const int* in_sizes, int n_in,
                              void* d_out, int out_size, void* d_ws, size_t ws_size,
                              hipStream_t stream) {
    // Input order per setup_inputs():
    const int*   seq          = (const int*)  d_in[0];   // [B*L]
    const float* atom_embed   = (const float*)d_in[2];
    const float* bond_embed   = (const float*)d_in[3];
    const float* ext_embed    = (const float*)d_in[4];
    const float* W_l2l        = (const float*)d_in[5];   // [3,32,32]
    const float* b_l2l        = (const float*)d_in[6];   // [3,32]
    const float* W_l2o        = (const float*)d_in[7];   // [3,32,32]
    const float* b_l2o        = (const float*)d_in[8];   // [3,32]
    const int*   edge_sym     = (const int*)  d_in[9];
    const int*   node_sym     = (const int*)  d_in[10];
    const int*   node_ext     = (const int*)  d_in[11];
    const float* node_has_ext = (const float*)d_in[12];
    const int*   edge_rule    = (const int*)  d_in[13];
    const int*   node_rule    = (const int*)  d_in[14];
    const int*   inc_edge     = (const int*)  d_in[15];
    const int*   inc_node     = (const int*)  d_in[16];

    const int  E   = in_sizes[9];
    const int  N   = in_sizes[10];
    const int  I2  = in_sizes[15];          // 2*E incidences
    const long BL  = (long)in_sizes[0];     // B*L

    // Workspace layout (256B aligned slabs)
    size_t off = 0;
    auto carve = [&](size_t bytes) {
        size_t o = off;
        off = (off + bytes + 255) & ~(size_t)255;
        return o;
    };
    char* ws = (char*)d_ws;
    float* fp  = (float*)(ws + carve((size_t)NRULES * DIM * sizeof(float)));
    float* h_e = (float*)(ws + carve((size_t)E * DIM * sizeof(float)));
    float* v_e = (float*)(ws + carve((size_t)E * DIM * sizeof(float)));
    float* h_n = (float*)(ws + carve((size_t)N * DIM * sizeof(float)));
    float* v_n = (float*)(ws + carve((size_t)N * DIM * sizeof(float)));

    const int TPB = 256;

    // 1) embeddings + zero fp
    long initT = (long)(E + N + NRULES) * DIM;
    init_kernel<<<dim3((unsigned)((initT + TPB - 1) / TPB)), dim3(TPB), 0, stream>>>(
        h_e, h_n, fp, atom_embed, bond_embed, ext_embed,
        edge_sym, node_sym, node_ext, node_has_ext, E, N);

    const int tilesE = (E + 15) / 16, blocksE = (tilesE + 3) / 4;
    const int tilesN = (N + 15) / 16, blocksN = (tilesN + 3) / 4;

    for (int l = 0; l < 3; ++l) {
        // 2) v = h
        long seedT = (long)(E + N) * (DIM / 4);
        seed_kernel<<<dim3((unsigned)((seedT + TPB - 1) / TPB)), dim3(TPB), 0, stream>>>(
            (float4*)v_e, (const float4*)h_e, (float4*)v_n, (const float4*)h_n, E, N);

        // 3) incidence scatter (segment sums)
        long scatT = (long)I2 * DIM;
        scatter_kernel<<<dim3((unsigned)((scatT + TPB - 1) / TPB)), dim3(TPB), 0, stream>>>(
            v_e, v_n, h_e, h_n, inc_edge, inc_node, I2);

        // 4) WMMA dual-GEMM: edges then nodes (each overwrites its h, adds to fp)
        const float* Wl = W_l2l + (size_t)l * DIM * DIM;
        const float* bl = b_l2l + (size_t)l * DIM;
        const float* Wo = W_l2o + (size_t)l * DIM * DIM;
        const float* bo = b_l2o + (size_t)l * DIM;
        tile_gemm_kernel<<<dim3((unsigned)blocksE), dim3(128), 0, stream>>>(
            v_e, h_e, fp, edge_rule, Wl, bl, Wo, bo, E);
        tile_gemm_kernel<<<dim3((unsigned)blocksN), dim3(128), 0, stream>>>(
            v_n, h_n, fp, node_rule, Wl, bl, Wo, bo, N);
    }

    // 5) final gather to [B,L,32]
    long gatT = BL * (DIM / 4);
    gather_kernel<<<dim3((unsigned)((gatT + TPB - 1) / TPB)), dim3(TPB), 0, stream>>>(
        seq, (const float4*)fp, (float4*)d_out, BL);
}